// emdModule_30975304139310
// MI455X (gfx1250) — compile-verified
//
#include <hip/hip_runtime.h>
#include <hip/hip_bf16.h>
#include <math.h>

#define B_     8
#define N_     2048
#define BN_    (B_ * N_)
#define ITERS_ 20

typedef float v2f __attribute__((ext_vector_type(2)));
typedef float v8f __attribute__((ext_vector_type(8)));

// Branchless combine of two (top1, argcol, top2) summaries over disjoint
// column sets; lowest column index wins ties (matches jnp.argmax).
// Non-short-circuit | & so the compiler emits v_cmp/v_cndmask, not branches.
__device__ __forceinline__ void top2_merge(float& a1, int& aj, float& a2,
                                           float b1, int bj, float b2) {
  bool take = (b1 > a1) | ((b1 == a1) & (bj < aj));
  float c1 = take ? a1 : b1;                 // loser's top1
  float m2a = take ? b2 : a2;                // winner's top2
  float m2 = (c1 > m2a) ? c1 : m2a;
  float m1 = take ? b1 : a1;
  int   mj = take ? bj : aj;
  a1 = m1; aj = mj; a2 = m2;
}

__global__ __launch_bounds__(256) void emd_init(
    const float* __restrict__ xyz1, const float* __restrict__ xyz2,
    float* price, int* assign, int* ainv, unsigned* maxbid, int* winner,
    float* sq1, float* sq2) {
  int idx = blockIdx.x * blockDim.x + threadIdx.x;
  if (idx >= BN_) return;
  price[idx]  = 0.0f;
  assign[idx] = -1;
  ainv[idx]   = -1;
  maxbid[idx] = 0u;
  winner[idx] = N_;
  float x = xyz1[idx * 3 + 0], y = xyz1[idx * 3 + 1], z = xyz1[idx * 3 + 2];
  sq1[idx] = x * x + y * y + z * z;
  x = xyz2[idx * 3 + 0]; y = xyz2[idx * 3 + 1]; z = xyz2[idx * 3 + 2];
  sq2[idx] = x * x + y * y + z * z;
}

// One block = (batch, 8 row-tiles). Each wave32 owns one 16-row tile and
// sweeps all 128 column tiles with V_WMMA_F32_16X16X4_F32 (K = {x,y,z,0}).
// Each lane keeps a private top-2 over its column residue class; one
// 16-lane butterfly per row AFTER the loop merges the classes.
__global__ __launch_bounds__(256) void emd_bid(
    const float* __restrict__ xyz1, const float* __restrict__ xyz2,
    const float* __restrict__ sq1, const float* __restrict__ sq2,
    const float* __restrict__ price, const int* __restrict__ assign,
    int* __restrict__ bid_j, float* __restrict__ bid_incr,
    unsigned* __restrict__ maxbid, const float* __restrict__ epsp) {
  __shared__ float sb0[2 * N_];   // [half][col]: half0 = x, half1 = z (B-frag K0/K2)
  __shared__ float sy[N_];        // B-frag K1 for half 0
  __shared__ float scp[N_];       // ||b||^2 + price
  const int b   = blockIdx.x >> 4;
  const int blk = blockIdx.x & 15;
  const float eps = *epsp;

  const float* x2 = xyz2 + (size_t)b * N_ * 3;
  for (int j = threadIdx.x; j < N_; j += 256) {
    sb0[j]      = x2[j * 3 + 0];
    sb0[N_ + j] = x2[j * 3 + 2];
    sy[j]       = x2[j * 3 + 1];
    scp[j]      = sq2[b * N_ + j] + price[b * N_ + j];
  }
  __syncthreads();

  const int wave = threadIdx.x >> 5;
  const int lane = threadIdx.x & 31;
  const int half = lane >> 4;                  // K-half for A/B fragments
  const int ln   = lane & 15;
  const int hbase = half * N_;
  const int rowbase = (blk * 8 + wave) * 16;   // 128 row tiles / batch

  // A fragment, 16x4 f32: lanes 0-15 hold (x,y) of row ln; lanes 16-31 hold (z,0)
  const float* x1p = xyz1 + ((size_t)(b * N_) + rowbase + ln) * 3;
  v2f afrag;
  afrag[0] = half ? x1p[2] : x1p[0];
  afrag[1] = half ? 0.0f   : x1p[1];

  float ns1[8];                                // -0.5 * ||a||^2 per C/D slot
#pragma unroll
  for (int v = 0; v < 8; ++v)
    ns1[v] = -0.5f * sq1[b * N_ + rowbase + v + 8 * half];

  // per-lane running top-2 over this lane's column residue class (d-scale)
  float r1[8], r2[8]; int rj[8];
#pragma unroll
  for (int v = 0; v < 8; ++v) { r1[v] = -INFINITY; r2[v] = -INFINITY; rj[v] = 0; }

#pragma unroll 2
  for (int ct = 0; ct < N_ / 16; ++ct) {
    const int c = ct * 16 + ln;                // this lane's column
    float y = sy[c];                           // unconditional load, cndmask below
    v2f bfrag;                                 // B fragment, 4x16 f32
    bfrag[0] = sb0[hbase + c];
    bfrag[1] = half ? 0.0f : y;
    const float ncp = -0.5f * scp[c];
    v8f cacc;
#pragma unroll
    for (int v = 0; v < 8; ++v) cacc[v] = ns1[v] + ncp;
    // D = a.b - (|a|^2 + |b|^2 + price)/2  ==  value / 2
    v8f d = __builtin_amdgcn_wmma_f32_16x16x4_f32(
        false, afrag, false, bfrag, (short)0, cacc, false, false);
#pragma unroll
    for (int v = 0; v < 8; ++v) {              // branchless top-2 insert
      float val = d[v];
      bool take = val > r1[v];                 // strict: ties keep lower col
      float mx2 = (val > r2[v]) ? val : r2[v]; // cmp+select: no NaN-canonicalize max
      float nr2 = take ? r1[v] : mx2;
      r1[v] = take ? val : r1[v];
      rj[v] = take ? c : rj[v];
      r2[v] = nr2;
    }
  }

  // one 16-lane butterfly per row merges the residue classes (runs once)
#pragma unroll
  for (int v = 0; v < 8; ++v) {
    float a1 = r1[v]; int aj = rj[v]; float a2 = r2[v];
#pragma unroll
    for (int mask = 1; mask <= 8; mask <<= 1) {
      float o1 = __shfl_xor(a1, mask, 32);
      int   oj = __shfl_xor(aj, mask, 32);
      float o2 = __shfl_xor(a2, mask, 32);
      top2_merge(a1, aj, a2, o1, oj, o2);
    }
    r1[v] = a1; rj[v] = aj; r2[v] = a2;
  }

  // one designated lane per row posts the bid
#pragma unroll
  for (int v = 0; v < 8; ++v) {
    if (ln == v) {
      const int row = rowbase + v + 8 * half;
      const int gi  = b * N_ + row;
      if (assign[gi] < 0) {
        float incr   = 2.0f * (r1[v] - r2[v]) + eps;   // back to value scale; > 0
        bid_j[gi]    = rj[v];
        bid_incr[gi] = incr;
        atomicMax(&maxbid[b * N_ + rj[v]], __float_as_uint(incr));
      } else {
        bid_j[gi] = -1;
      }
    }
  }
}

__global__ __launch_bounds__(256) void emd_match(
    const int* __restrict__ bid_j, const float* __restrict__ bid_incr,
    const unsigned* __restrict__ maxbid, int* __restrict__ winner) {
  int idx = blockIdx.x * blockDim.x + threadIdx.x;
  if (idx >= BN_) return;
  int b = idx >> 11, row = idx & (N_ - 1);
  int j = bid_j[idx];
  if (j >= 0 && __float_as_uint(bid_incr[idx]) == maxbid[b * N_ + j])
    atomicMin(&winner[b * N_ + j], row);            // lowest-index bidder wins
}

__global__ __launch_bounds__(256) void emd_assign(
    int* __restrict__ assign, int* __restrict__ ainv,
    unsigned* __restrict__ maxbid, int* __restrict__ winner,
    float* __restrict__ price) {
  int idx = blockIdx.x * blockDim.x + threadIdx.x;
  if (idx >= BN_) return;
  int b = idx >> 11, j = idx & (N_ - 1);
  int w = winner[idx];
  if (w < N_) {
    int prev = ainv[idx];
    if (prev >= 0) assign[b * N_ + prev] = -1;      // prev-holders and winners are disjoint
    assign[b * N_ + w] = j;
    ainv[idx]  = w;
    price[idx] += __uint_as_float(maxbid[idx]);
  }
  maxbid[idx] = 0u;                                  // reset for next round
  winner[idx] = N_;
}

__global__ __launch_bounds__(256) void emd_final(
    const float* __restrict__ xyz1, const float* __restrict__ xyz2,
    const int* __restrict__ assign, float* __restrict__ dist_out,
    int* __restrict__ asgn_out) {
  int idx = blockIdx.x * blockDim.x + threadIdx.x;
  if (idx >= BN_) return;
  int b = idx >> 11;
  int a = assign[idx];
  float d = 0.0f;
  if (a >= 0) {
    const float* p1 = xyz1 + (size_t)idx * 3;
    const float* p2 = xyz2 + ((size_t)(b * N_) + a) * 3;
    float dx = p1[0] - p2[0], dy = p1[1] - p2[1], dz = p1[2] - p2[2];
    d = dx * dx + dy * dy + dz * dz;
  }
  dist_out[idx] = d;
  asgn_out[idx] = a;
}

extern "C" void kernel_launch(void* const* d_in, const int* in_sizes, int n_in,
                              void* d_out, int out_size, void* d_ws, size_t ws_size,
                              hipStream_t stream) {
  const float* xyz1 = (const float*)d_in[0];
  const float* xyz2 = (const float*)d_in[1];
  const float* epsp = (const float*)d_in[2];
  (void)in_sizes; (void)n_in; (void)out_size; (void)ws_size;

  char* ws = (char*)d_ws;
  float*    price    = (float*)   (ws + 0 * BN_ * 4);
  int*      assign   = (int*)     (ws + 1 * BN_ * 4);
  int*      ainv     = (int*)     (ws + 2 * BN_ * 4);
  unsigned* maxbid   = (unsigned*)(ws + 3 * BN_ * 4);
  int*      winner   = (int*)     (ws + 4 * BN_ * 4);
  int*      bid_j    = (int*)     (ws + 5 * BN_ * 4);
  float*    bid_incr = (float*)   (ws + 6 * BN_ * 4);
  float*    sq1      = (float*)   (ws + 7 * BN_ * 4);
  float*    sq2      = (float*)   (ws + 8 * BN_ * 4);

  const int tpb = 256;
  const int gBN = BN_ / tpb;

  emd_init<<<gBN, tpb, 0, stream>>>(xyz1, xyz2, price, assign, ainv, maxbid,
                                    winner, sq1, sq2);
  for (int it = 0; it < ITERS_; ++it) {
    emd_bid<<<B_ * 16, tpb, 0, stream>>>(xyz1, xyz2, sq1, sq2, price, assign,
                                         bid_j, bid_incr, maxbid, epsp);
    emd_match<<<gBN, tpb, 0, stream>>>(bid_j, bid_incr, maxbid, winner);
    emd_assign<<<gBN, tpb, 0, stream>>>(assign, ainv, maxbid, winner, price);
  }
  emd_final<<<gBN, tpb, 0, stream>>>(xyz1, xyz2, assign, (float*)d_out,
                                     (int*)d_out + BN_);
}